// GS_712964571540
// MI455X (gfx1250) — compile-verified
//
#include <hip/hip_runtime.h>

// MI455X / gfx1250, wave32. fp32 WMMA path: V_WMMA_F32_16X16X4_F32.
typedef float v2f __attribute__((ext_vector_type(2)));
typedef float v8f __attribute__((ext_vector_type(8)));

#define N_DIM   2048
#define BS_DIM  64
#define N_ITER  25
#define ELEMS   ((size_t)BS_DIM * N_DIM)      // 131072
#define MATSZ   ((size_t)N_DIM * N_DIM)       // 4194304
#define KSPLIT  4
#define KSLICE  (N_DIM / KSPLIT)              // 512

// ---------------------------------------------------------------------------
// traj[0] must be exactly zero (harness poisons d_out with 0xAA).
__global__ void gs_zero_kernel(float* __restrict__ p, int n) {
    int i = blockIdx.x * blockDim.x + threadIdx.x;
    if (i < n) p[i] = 0.0f;
}

// ---------------------------------------------------------------------------
// LDS-tiled 2048x2048 fp32 transpose: out[n][k] = in[k][n].
__global__ void __launch_bounds__(256)
gs_transpose(const float* __restrict__ in, float* __restrict__ out) {
    __shared__ float tile[32][33];
    const int bx = blockIdx.x & 63;
    const int by = blockIdx.x >> 6;
    const int tx = threadIdx.x & 31;
    const int ty = threadIdx.x >> 5;

    const int x = bx * 32 + tx;
    #pragma unroll
    for (int r = ty; r < 32; r += 8)
        tile[r][tx] = in[(size_t)(by * 32 + r) * N_DIM + x];
    __syncthreads();
    const int x2 = by * 32 + tx;
    #pragma unroll
    for (int r = ty; r < 32; r += 8)
        out[(size_t)(bx * 32 + r) * N_DIM + x2] = tile[tx][r];
}

// ---------------------------------------------------------------------------
// Split-K partial GEMM: P[s] = A[64,2048] @ Bt^T over K-slice s (K=512).
// grid = 512 blocks x 128 thr: block -> (kslice s, N-tile tn); wave -> M-tile.
// 4 accumulator chains so dependent WMMAs are 4 issues apart (XDL latency).
// Both fragments are contiguous b64 loads.
__global__ void __launch_bounds__(128)
gs_gemm_ksplit(const float* __restrict__ A, const float* __restrict__ Bt,
               float* __restrict__ P) {
    const int lane = threadIdx.x & 31;
    const int tm   = threadIdx.x >> 5;        // 4 M-tiles (M=64)
    const int s    = blockIdx.x & 3;          // K-slice
    const int tn   = blockIdx.x >> 2;         // 128 N-tiles
    const int l15  = lane & 15;
    const int hi   = lane >> 4;               // 0: K=0,1  1: K=2,3 (ISA layout)
    const int kbase = s * KSLICE + hi * 2;

    const float* aptr = A  + (size_t)(tm * 16 + l15) * N_DIM + kbase;
    const float* bptr = Bt + (size_t)(tn * 16 + l15) * N_DIM + kbase;

    v8f c0 = {}; v8f c1 = {}; v8f c2 = {}; v8f c3 = {};
    #pragma unroll 2
    for (int k = 0; k < KSLICE; k += 16) {
        v2f a0 = *(const v2f*)(aptr + k);
        v2f b0 = *(const v2f*)(bptr + k);
        v2f a1 = *(const v2f*)(aptr + k + 4);
        v2f b1 = *(const v2f*)(bptr + k + 4);
        v2f a2 = *(const v2f*)(aptr + k + 8);
        v2f b2 = *(const v2f*)(bptr + k + 8);
        v2f a3 = *(const v2f*)(aptr + k + 12);
        v2f b3 = *(const v2f*)(bptr + k + 12);
        c0 = __builtin_amdgcn_wmma_f32_16x16x4_f32(false, a0, false, b0,
                                                   (short)0, c0, false, false);
        c1 = __builtin_amdgcn_wmma_f32_16x16x4_f32(false, a1, false, b1,
                                                   (short)0, c1, false, false);
        c2 = __builtin_amdgcn_wmma_f32_16x16x4_f32(false, a2, false, b2,
                                                   (short)0, c2, false, false);
        c3 = __builtin_amdgcn_wmma_f32_16x16x4_f32(false, a3, false, b3,
                                                   (short)0, c3, false, false);
    }

    // C/D layout: VGPR r -> row M = r + 8*hi, col = lane&15
    float* optr = P + (size_t)s * ELEMS
                    + (size_t)(tm * 16 + hi * 8) * N_DIM + tn * 16 + l15;
    #pragma unroll
    for (int r = 0; r < 8; ++r)
        optr[(size_t)r * N_DIM] = (c0[r] + c1[r]) + (c2[r] + c3[r]);
}

// ---------------------------------------------------------------------------
// Deterministic split-K reduction + fused epilogue:
// v = P0+P1+P2+P3 (fixed order); res = subSrc ? subSrc - v : v;
// res -> out0 (+ optional out1/out2). float4 per thread.
__global__ void __launch_bounds__(256)
gs_reduce(const float* __restrict__ P, const float* __restrict__ subSrc,
          float* __restrict__ out0, float* __restrict__ out1,
          float* __restrict__ out2) {
    const size_t i = ((size_t)blockIdx.x * 256 + threadIdx.x) * 4;
    float4 v0 = *(const float4*)(P + i);
    float4 v1 = *(const float4*)(P + ELEMS + i);
    float4 v2 = *(const float4*)(P + 2 * ELEMS + i);
    float4 v3 = *(const float4*)(P + 3 * ELEMS + i);
    float4 v;
    v.x = (v0.x + v1.x) + (v2.x + v3.x);
    v.y = (v0.y + v1.y) + (v2.y + v3.y);
    v.z = (v0.z + v1.z) + (v2.z + v3.z);
    v.w = (v0.w + v1.w) + (v2.w + v3.w);
    if (subSrc) {
        float4 s = *(const float4*)(subSrc + i);
        v.x = s.x - v.x; v.y = s.y - v.y; v.z = s.z - v.z; v.w = s.w - v.w;
    }
    *(float4*)(out0 + i) = v;
    if (out1) *(float4*)(out1 + i) = v;
    if (out2) *(float4*)(out2 + i) = v;
}

// ---------------------------------------------------------------------------
// Fallback (small workspace): fused non-split GEMMs from the previous round.
__global__ void __launch_bounds__(128)
gs_gemm_nt(const float* __restrict__ A, const float* __restrict__ Bt,
           const float* __restrict__ subSrc,
           float* __restrict__ out0, float* __restrict__ out1,
           float* __restrict__ out2) {
    const int lane = threadIdx.x & 31;
    const int wave = threadIdx.x >> 5;
    const int t    = blockIdx.x * 4 + wave;
    const int tm   = t & 3;
    const int tn   = t >> 2;
    const int l15  = lane & 15;
    const int hi   = lane >> 4;
    const int koff = hi * 2;

    const float* aptr = A  + (size_t)(tm * 16 + l15) * N_DIM + koff;
    const float* bptr = Bt + (size_t)(tn * 16 + l15) * N_DIM + koff;

    v8f acc0 = {}; v8f acc1 = {};
    #pragma unroll 4
    for (int k = 0; k < N_DIM; k += 8) {
        v2f a0 = *(const v2f*)(aptr + k);
        v2f b0 = *(const v2f*)(bptr + k);
        v2f a1 = *(const v2f*)(aptr + k + 4);
        v2f b1 = *(const v2f*)(bptr + k + 4);
        acc0 = __builtin_amdgcn_wmma_f32_16x16x4_f32(false, a0, false, b0,
                                                     (short)0, acc0, false, false);
        acc1 = __builtin_amdgcn_wmma_f32_16x16x4_f32(false, a1, false, b1,
                                                     (short)0, acc1, false, false);
    }

    const size_t obase = (size_t)(tm * 16 + hi * 8) * N_DIM + tn * 16 + l15;
    float res[8];
    #pragma unroll
    for (int r = 0; r < 8; ++r) res[r] = acc0[r] + acc1[r];
    if (subSrc) {
        #pragma unroll
        for (int r = 0; r < 8; ++r)
            res[r] = subSrc[obase + (size_t)r * N_DIM] - res[r];
    }
    #pragma unroll
    for (int r = 0; r < 8; ++r) out0[obase + (size_t)r * N_DIM] = res[r];
    if (out1) {
        #pragma unroll
        for (int r = 0; r < 8; ++r) out1[obase + (size_t)r * N_DIM] = res[r];
    }
    if (out2) {
        #pragma unroll
        for (int r = 0; r < 8; ++r) out2[obase + (size_t)r * N_DIM] = res[r];
    }
}

__global__ void __launch_bounds__(128)
gs_gemm_nn(const float* __restrict__ A, const float* __restrict__ B,
           const float* __restrict__ subSrc,
           float* __restrict__ out0, float* __restrict__ out1,
           float* __restrict__ out2) {
    const int lane = threadIdx.x & 31;
    const int wave = threadIdx.x >> 5;
    const int t    = blockIdx.x * 4 + wave;
    const int tm   = t & 3;
    const int tn   = t >> 2;
    const int l15  = lane & 15;
    const int hi   = lane >> 4;
    const int koff = hi * 2;

    const float* aptr = A + (size_t)(tm * 16 + l15) * N_DIM + koff;
    const float* bptr = B + (size_t)koff * N_DIM + tn * 16 + l15;

    v8f acc0 = {}; v8f acc1 = {};
    #pragma unroll 4
    for (int k = 0; k < N_DIM; k += 8) {
        v2f a0 = *(const v2f*)(aptr + k);
        v2f b0;
        b0.x = bptr[(size_t)(k + 0) * N_DIM];
        b0.y = bptr[(size_t)(k + 1) * N_DIM];
        v2f a1 = *(const v2f*)(aptr + k + 4);
        v2f b1;
        b1.x = bptr[(size_t)(k + 4) * N_DIM];
        b1.y = bptr[(size_t)(k + 5) * N_DIM];
        acc0 = __builtin_amdgcn_wmma_f32_16x16x4_f32(false, a0, false, b0,
                                                     (short)0, acc0, false, false);
        acc1 = __builtin_amdgcn_wmma_f32_16x16x4_f32(false, a1, false, b1,
                                                     (short)0, acc1, false, false);
    }

    const size_t obase = (size_t)(tm * 16 + hi * 8) * N_DIM + tn * 16 + l15;
    float res[8];
    #pragma unroll
    for (int r = 0; r < 8; ++r) res[r] = acc0[r] + acc1[r];
    if (subSrc) {
        #pragma unroll
        for (int r = 0; r < 8; ++r)
            res[r] = subSrc[obase + (size_t)r * N_DIM] - res[r];
    }
    #pragma unroll
    for (int r = 0; r < 8; ++r) out0[obase + (size_t)r * N_DIM] = res[r];
    if (out1) {
        #pragma unroll
        for (int r = 0; r < 8; ++r) out1[obase + (size_t)r * N_DIM] = res[r];
    }
    if (out2) {
        #pragma unroll
        for (int r = 0; r < 8; ++r) out2[obase + (size_t)r * N_DIM] = res[r];
    }
}

// ---------------------------------------------------------------------------
extern "C" void kernel_launch(void* const* d_in, const int* in_sizes, int n_in,
                              void* d_out, int out_size, void* d_ws, size_t ws_size,
                              hipStream_t stream) {
    // inputs: num_itr, bs, y, H, Dinv, U, invM (num_itr/bs known at compile time)
    const float* y    = (const float*)d_in[2];
    const float* H    = (const float*)d_in[3];
    const float* Dinv = (const float*)d_in[4];
    const float* U    = (const float*)d_in[5];
    const float* invM = (const float*)d_in[6];

    float* sfinal = (float*)d_out;                 // [64,2048]
    float* traj   = (float*)d_out + ELEMS;         // [26,64,2048]

    float* ws    = (float*)d_ws;
    float* yMF   = ws;                             // [64,2048]
    float* S     = ws + ELEMS;                     // [64,2048]
    float* T     = ws + 2 * ELEMS;                 // [64,2048]
    float* P     = ws + 3 * ELEMS;                 // [4,64,2048] split-K partials
    float* Dt    = ws + 7 * ELEMS;                 // Dinv^T [2048,2048]
    float* Ut    = Dt + MATSZ;                     // U^T
    float* invMt = Ut + MATSZ;                     // invM^T

    const size_t need_fast = (7 * ELEMS + 3 * MATSZ) * sizeof(float); // ~54 MB
    const size_t need_mid  = (3 * ELEMS + 2 * MATSZ) * sizeof(float); // ~35 MB
    const int mode = (ws_size >= need_fast) ? 2 : (ws_size >= need_mid ? 1 : 0);

    dim3 gblk(128);
    dim3 ggrd(128);                 // non-split: 512 tiles
    dim3 kgrd(512);                 // split-K: 4 slices x 128 N-tiles, 2048 waves
    dim3 rgrd(ELEMS / (256 * 4));   // reduce: 128 blocks

    // traj[0] = zeros
    gs_zero_kernel<<<(int)(ELEMS / 256), 256, 0, stream>>>(traj, (int)ELEMS);

    if (mode == 2) {
        // One-time transposes; all iteration GEMMs then use contiguous b64 frags.
        gs_transpose<<<4096, 256, 0, stream>>>(Dinv, Dt);
        gs_transpose<<<4096, 256, 0, stream>>>(U, Ut);
        gs_transpose<<<4096, 256, 0, stream>>>(invM, invMt);

        // yMF = y @ H^T
        gs_gemm_ksplit<<<kgrd, gblk, 0, stream>>>(y, H, P);
        gs_reduce<<<rgrd, 256, 0, stream>>>(P, nullptr, yMF, nullptr, nullptr);
        // s_init = yMF @ Dinv
        gs_gemm_ksplit<<<kgrd, gblk, 0, stream>>>(yMF, Dt, P);
        gs_reduce<<<rgrd, 256, 0, stream>>>(P, nullptr, S, nullptr, nullptr);

        for (int i = 1; i <= N_ITER; ++i) {
            float* trajD  = traj + (size_t)i * ELEMS;
            float* finalD = (i == N_ITER) ? sfinal : nullptr;
            // T = yMF - S @ U
            gs_gemm_ksplit<<<kgrd, gblk, 0, stream>>>(S, Ut, P);
            gs_reduce<<<rgrd, 256, 0, stream>>>(P, yMF, T, nullptr, nullptr);
            // S = T @ invM ; traj[i] ; s_final on last iter
            gs_gemm_ksplit<<<kgrd, gblk, 0, stream>>>(T, invMt, P);
            gs_reduce<<<rgrd, 256, 0, stream>>>(P, nullptr, S, trajD, finalD);
        }
    } else if (mode == 1) {
        gs_transpose<<<4096, 256, 0, stream>>>(U, Ut - MATSZ);      // reuse Dt slot
        gs_transpose<<<4096, 256, 0, stream>>>(invM, invMt - MATSZ);
        float* Ut1    = Dt;          // U^T at first matrix slot
        float* invMt1 = Dt + MATSZ;
        gs_gemm_nt<<<ggrd, gblk, 0, stream>>>(y, H, nullptr, yMF, nullptr, nullptr);
        gs_gemm_nn<<<ggrd, gblk, 0, stream>>>(yMF, Dinv, nullptr, S, nullptr, nullptr);
        for (int i = 1; i <= N_ITER; ++i) {
            float* trajD  = traj + (size_t)i * ELEMS;
            float* finalD = (i == N_ITER) ? sfinal : nullptr;
            gs_gemm_nt<<<ggrd, gblk, 0, stream>>>(S, Ut1, yMF, T, nullptr, nullptr);
            gs_gemm_nt<<<ggrd, gblk, 0, stream>>>(T, invMt1, nullptr, S, trajD, finalD);
        }
    } else {
        gs_gemm_nt<<<ggrd, gblk, 0, stream>>>(y, H, nullptr, yMF, nullptr, nullptr);
        gs_gemm_nn<<<ggrd, gblk, 0, stream>>>(yMF, Dinv, nullptr, S, nullptr, nullptr);
        for (int i = 1; i <= N_ITER; ++i) {
            float* trajD  = traj + (size_t)i * ELEMS;
            float* finalD = (i == N_ITER) ? sfinal : nullptr;
            gs_gemm_nn<<<ggrd, gblk, 0, stream>>>(S, U, yMF, T, nullptr, nullptr);
            gs_gemm_nn<<<ggrd, gblk, 0, stream>>>(T, invM, nullptr, S, trajD, finalD);
        }
    }
}